// GINLayer_55783035240590
// MI455X (gfx1250) — compile-verified
//
#include <hip/hip_runtime.h>

// GIN layer for MI455X (gfx1250, wave32):
//   agg = x + segment_sum(x[src], dst)          -> atomics into d_out (L2-resident)
//   h1  = agg @ W1 + b1  (+ column sum/sumsq)   -> WMMA f32 16x16x4, stats via atomics
//   scale/shift from BN stats                   -> tiny kernel
//   out = relu( relu(BN(h1)) @ W2 + b2 )        -> WMMA f32 16x16x4, BN+ReLU fused in A-load

typedef __attribute__((ext_vector_type(2))) float v2f;
typedef __attribute__((ext_vector_type(4))) float v4f;
typedef __attribute__((ext_vector_type(8))) float v8f;

#define D 64
#define LDSW 66   // padded LDS stride: 66*4 bytes keeps b64 loads 8B-aligned, avoids conflicts
#define BN_EPS 1e-5f

// ---------------------------------------------------------------- init: agg = x
__global__ void gin_init_agg(const float* __restrict__ x, float* __restrict__ agg, int total4) {
    int i = blockIdx.x * blockDim.x + threadIdx.x;
    int stride = gridDim.x * blockDim.x;
    for (; i < total4; i += stride) {
        ((v4f*)agg)[i] = ((const v4f*)x)[i];
    }
}

__global__ void gin_zero_stats(float* __restrict__ stats) {
    if (threadIdx.x < 128) stats[threadIdx.x] = 0.0f;
}

// ------------------------------------------------- scatter: agg[dst] += x[src]
// one thread handles 4 consecutive features of one edge (16 threads/edge)
__global__ void gin_scatter(const float* __restrict__ x, const int* __restrict__ src,
                            const int* __restrict__ dst, float* __restrict__ agg, int E) {
    int total = E * 16;
    int i = blockIdx.x * blockDim.x + threadIdx.x;
    int stride = gridDim.x * blockDim.x;
    for (; i < total; i += stride) {
        int e = i >> 4;
        int f = (i & 15) << 2;
        int s = src[e];
        int d = dst[e];
        v4f v = *(const v4f*)(x + (long long)s * D + f);
        float* p = agg + (long long)d * D + f;
        atomicAdd(p + 0, v.x);
        atomicAdd(p + 1, v.y);
        atomicAdd(p + 2, v.z);
        atomicAdd(p + 3, v.w);
    }
}

// --------------------------------------- GEMM1: h1 = A@W1 + b1, accumulate BN stats
__global__ void __launch_bounds__(128)
gin_gemm1(const float* __restrict__ A, const float* __restrict__ W,
          const float* __restrict__ bias, float* __restrict__ out,
          float* __restrict__ stats, int N) {
    __shared__ __align__(16) float Wt[D * LDSW];   // W transposed: Wt[n*LDSW + k] = W[k][n]
    const int tid = threadIdx.x;
    for (int idx = tid; idx < D * D; idx += 128) {
        int k = idx >> 6, n = idx & 63;
        Wt[n * LDSW + k] = W[idx];
    }
    __syncthreads();

    const int lane = tid & 31;
    const int wave = tid >> 5;
    const int m    = lane & 15;   // A-row / B-col slot within tile
    const int half = lane >> 4;   // K-pair selector (ISA 16x4 f32 A layout)
    const int N0   = wave * 16;   // output column tile (4 waves cover 64 cols)
    const int row0 = blockIdx.x * 16;

    const int arow = (row0 + m < N) ? (row0 + m) : (N - 1);
    const float* Arow = A + (long long)arow * D;
    const float* Brow = &Wt[(N0 + m) * LDSW];

    v8f acc = {0.f, 0.f, 0.f, 0.f, 0.f, 0.f, 0.f, 0.f};
#pragma unroll
    for (int kk = 0; kk < D; kk += 4) {
        const int k0 = kk + 2 * half;
        v2f a = *(const v2f*)(Arow + k0);               // A[m][k0], A[m][k0+1]
        v2f b = *(const v2f*)(Brow + k0);               // W[k0][N0+m], W[k0+1][N0+m]
        acc = __builtin_amdgcn_wmma_f32_16x16x4_f32(false, a, false, b,
                                                    (short)0, acc, false, false);
    }

    const float bcol = bias[N0 + m];
    float s = 0.f, s2 = 0.f;
#pragma unroll
    for (int r = 0; r < 8; ++r) {
        int row = row0 + r + 8 * half;                  // C/D layout: VGPR r -> row r+8*half
        float v = acc[r] + bcol;
        if (row < N) {
            out[(long long)row * D + N0 + m] = v;
            s  += v;
            s2 += v * v;
        }
    }
    atomicAdd(&stats[N0 + m], s);
    atomicAdd(&stats[64 + N0 + m], s2);
}

// ------------------------------------- stats -> (scale, shift) for fused BN
__global__ void gin_bn_finalize(float* __restrict__ stats, const float* __restrict__ gamma,
                                const float* __restrict__ beta, float invN) {
    int c = threadIdx.x;
    if (c < D) {
        float mean = stats[c] * invN;
        float var  = stats[64 + c] * invN - mean * mean;
        float sc   = gamma[c] * rsqrtf(var + BN_EPS);
        float sh   = beta[c] - mean * sc;
        stats[c]      = sc;
        stats[64 + c] = sh;
    }
}

// ---------------- GEMM2: out = relu( relu(h1*scale+shift) @ W2 + b2 ), BN fused in A
__global__ void __launch_bounds__(128)
gin_gemm2(const float* __restrict__ A, const float* __restrict__ W,
          const float* __restrict__ bias, const float* __restrict__ stats,
          float* __restrict__ out, int N) {
    __shared__ __align__(16) float Wt[D * LDSW];
    __shared__ __align__(16) float sc[D];
    __shared__ __align__(16) float sh[D];
    const int tid = threadIdx.x;
    for (int idx = tid; idx < D * D; idx += 128) {
        int k = idx >> 6, n = idx & 63;
        Wt[n * LDSW + k] = W[idx];
    }
    if (tid < D) { sc[tid] = stats[tid]; sh[tid] = stats[64 + tid]; }
    __syncthreads();

    const int lane = tid & 31;
    const int wave = tid >> 5;
    const int m    = lane & 15;
    const int half = lane >> 4;
    const int N0   = wave * 16;
    const int row0 = blockIdx.x * 16;

    const int arow = (row0 + m < N) ? (row0 + m) : (N - 1);
    const float* Arow = A + (long long)arow * D;
    const float* Brow = &Wt[(N0 + m) * LDSW];

    v8f acc = {0.f, 0.f, 0.f, 0.f, 0.f, 0.f, 0.f, 0.f};
#pragma unroll
    for (int kk = 0; kk < D; kk += 4) {
        const int k0 = kk + 2 * half;
        v2f a = *(const v2f*)(Arow + k0);
        v2f s = *(const v2f*)(sc + k0);
        v2f t = *(const v2f*)(sh + k0);
        a.x = fmaxf(a.x * s.x + t.x, 0.f);              // BN + ReLU fused
        a.y = fmaxf(a.y * s.y + t.y, 0.f);
        v2f b = *(const v2f*)(Brow + k0);
        acc = __builtin_amdgcn_wmma_f32_16x16x4_f32(false, a, false, b,
                                                    (short)0, acc, false, false);
    }

    const float bcol = bias[N0 + m];
#pragma unroll
    for (int r = 0; r < 8; ++r) {
        int row = row0 + r + 8 * half;
        if (row < N) {
            float v = acc[r] + bcol;
            out[(long long)row * D + N0 + m] = fmaxf(v, 0.f);  // final ReLU
        }
    }
}

extern "C" void kernel_launch(void* const* d_in, const int* in_sizes, int n_in,
                              void* d_out, int out_size, void* d_ws, size_t ws_size,
                              hipStream_t stream) {
    const float* x     = (const float*)d_in[0];
    const int*   ei    = (const int*)d_in[1];   // edge_index, int per harness convention
    const float* W1    = (const float*)d_in[2];
    const float* b1    = (const float*)d_in[3];
    const float* gamma = (const float*)d_in[4];
    const float* beta  = (const float*)d_in[5];
    const float* W2    = (const float*)d_in[6];
    const float* b2    = (const float*)d_in[7];
    float* out = (float*)d_out;

    const int N = in_sizes[0] / D;
    const int E = in_sizes[1] / 2;
    const int* src = ei;
    const int* dst = ei + E;

    // workspace: h1 (N*D floats) + BN stats (128 floats)
    float* h1    = (float*)d_ws;
    float* stats = h1 + (size_t)N * D;

    // 1) d_out <- x   (d_out doubles as the aggregation buffer)
    {
        int total4 = (N * D) / 4;
        int blocks = (total4 + 255) / 256;
        gin_init_agg<<<blocks, 256, 0, stream>>>(x, out, total4);
        gin_zero_stats<<<1, 128, 0, stream>>>(stats);
    }
    // 2) scatter-add neighbor features
    {
        long long total = (long long)E * 16;
        int blocks = (int)((total + 255) / 256);
        if (blocks > 131072) blocks = 131072;   // grid-stride handles the rest
        gin_scatter<<<blocks, 256, 0, stream>>>(x, src, dst, out, E);
    }
    // 3) h1 = agg@W1 + b1, column stats
    {
        int blocks = (N + 15) / 16;
        gin_gemm1<<<blocks, 128, 0, stream>>>(out, W1, b1, h1, stats, N);
    }
    // 4) BN scale/shift
    gin_bn_finalize<<<1, 64, 0, stream>>>(stats, gamma, beta, 1.0f / (float)N);
    // 5) out = relu( relu(BN(h1)) @ W2 + b2 )
    {
        int blocks = (N + 15) / 16;
        gin_gemm2<<<blocks, 128, 0, stream>>>(h1, W2, b2, stats, out, N);
    }
}